// GraphSAGE_52020643889765
// MI455X (gfx1250) — compile-verified
//
#include <hip/hip_runtime.h>
#include <hip/hip_bf16.h>
#include <math.h>

// GraphSAGE fused inference kernel for gfx1250 (MI455X).
// Live dataflow only (agg_embed_1 is dead in the reference):
//   X = [gather(nodes) | mean_{S} gather(neighbors_1)]   : [B, 2D]
//   Y = relu(X @ W2^T)                                   : [B, H]
//   P = softmax(Y @ Wo^T)                                : [B, O]
// fp32 end-to-end via V_WMMA_F32_16X16X4_F32 (matches reference precision;
// kernel is gather-bandwidth bound, so f32 WMMA is nowhere near the ceiling).

typedef __attribute__((ext_vector_type(2))) float v2f;
typedef __attribute__((ext_vector_type(8))) float v8f;

namespace sage {
constexpr int D  = 128;   // feature dim
constexpr int D2 = 256;   // concat dim
constexpr int H  = 128;   // hidden
constexpr int O  = 64;    // classes
constexpr int S  = 16;    // neighbor fan-out
constexpr int TM = 16;    // seed rows per block

// LDS strides padded so row bases stay 8-byte aligned (stride*4 % 8 == 0)
// while skewing bank access for the 16-lanes-read-16-rows WMMA A-fetch.
constexpr int XS = D2 + 2;  // 258 -> 1032 B/row
constexpr int YS = H + 2;   // 130 -> 520 B/row
constexpr int LS = O + 2;   // 66  -> 264 B/row
}

__global__ __launch_bounds__(128) void
graphsage_fused_kernel(const int* __restrict__ nodes,
                       const int* __restrict__ neigh1,
                       const float* __restrict__ feats,
                       const float* __restrict__ W2,
                       const float* __restrict__ Wo,
                       float* __restrict__ out)
{
    using namespace sage;
    __shared__ float Xs[TM * XS];   // 16.1 KB  concat input tile
    __shared__ float Ys[TM * YS];   //  8.1 KB  relu hidden tile
    __shared__ float Lg[TM * LS];   //  4.1 KB  logits tile

    const int row0 = blockIdx.x * TM;
    const int tid  = threadIdx.x;      // 0..127
    const int lane = tid & 31;
    const int wave = tid >> 5;         // 0..3

    // Warm L2/L1 with weight bytes this block will stream (global_prefetch_b8).
    __builtin_prefetch(W2 + (size_t)tid * (H * D2 / 128), 0, 0);
    __builtin_prefetch(Wo + (size_t)tid * (O * H / 128), 0, 0);

    // ---- Phase 0: gather origin features + neighbor mean into LDS ----------
    {
        const int c = tid;  // one column per thread, coalesced across the block
        for (int m = 0; m < TM; ++m) {
            const int nid = nodes[row0 + m];
            Xs[m * XS + c] = feats[(size_t)nid * D + c];
            const int* nb = neigh1 + (size_t)(row0 + m) * S;
            float acc = 0.0f;
            #pragma unroll 4
            for (int s = 0; s < S; ++s)
                acc += feats[(size_t)nb[s] * D + c];
            Xs[m * XS + D + c] = acc * (1.0f / (float)S);
        }
    }
    __syncthreads();

    // Lane roles for V_WMMA_F32_16X16X4_F32 (ISA §7.12.2):
    //   A (16x4): lane l -> row M=l&15, K = 2*(l>>4) + {0,1}
    //   B (4x16): lane l -> col N=l&15, K = 2*(l>>4) + {0,1}
    //   C (16x16): vgpr v, lanes 0-15 -> M=v, lanes 16-31 -> M=v+8, N=l&15
    const int mrow  = lane & 15;
    const int khalf = (lane >> 4) * 2;
    const int crow0 = (lane >> 4) * 8;

    // ---- Layer 1: Y = relu(X @ W2^T); 8 N-tiles, 2 per wave, K=256 --------
    for (int nt = wave; nt < H / 16; nt += 4) {
        v8f acc = {};
        const float* __restrict__ w2row =
            W2 + (size_t)(nt * 16 + mrow) * D2 + khalf;
        const float* __restrict__ xrow = &Xs[mrow * XS + khalf];
        #pragma unroll 8
        for (int k = 0; k < D2; k += 4) {
            v2f a = *(const v2f*)(xrow + k);    // LDS  ds_load_b64
            v2f b = *(const v2f*)(w2row + k);   // VMEM global_load_b64
            acc = __builtin_amdgcn_wmma_f32_16x16x4_f32(
                      false, a, false, b, (short)0, acc, false, false);
        }
        #pragma unroll
        for (int v = 0; v < 8; ++v)
            Ys[(crow0 + v) * YS + nt * 16 + mrow] = fmaxf(acc[v], 0.0f);
    }
    __syncthreads();

    // ---- Layer 2: logits = Y @ Wo^T; 4 N-tiles, 1 per wave, K=128 ---------
    {
        const int nt = wave;
        v8f acc = {};
        const float* __restrict__ worow =
            Wo + (size_t)(nt * 16 + mrow) * H + khalf;
        const float* __restrict__ yrow = &Ys[mrow * YS + khalf];
        #pragma unroll 8
        for (int k = 0; k < H; k += 4) {
            v2f a = *(const v2f*)(yrow + k);
            v2f b = *(const v2f*)(worow + k);
            acc = __builtin_amdgcn_wmma_f32_16x16x4_f32(
                      false, a, false, b, (short)0, acc, false, false);
        }
        #pragma unroll
        for (int v = 0; v < 8; ++v)
            Lg[(crow0 + v) * LS + nt * 16 + mrow] = acc[v];
    }
    __syncthreads();

    // ---- Softmax over O=64, one row per thread (threads 0..15) ------------
    if (tid < TM) {
        const int m = tid;
        float mx = -INFINITY;
        for (int j = 0; j < O; ++j) mx = fmaxf(mx, Lg[m * LS + j]);
        float sum = 0.0f;
        for (int j = 0; j < O; ++j) {
            float e = __expf(Lg[m * LS + j] - mx);
            Lg[m * LS + j] = e;            // reuse LDS, avoid 64-reg spill
            sum += e;
        }
        const float inv = 1.0f / sum;
        float* __restrict__ orow = out + (size_t)(row0 + m) * O;
        for (int j = 0; j < O; ++j) orow[j] = Lg[m * LS + j] * inv;
    }
}

extern "C" void kernel_launch(void* const* d_in, const int* in_sizes, int n_in,
                              void* d_out, int out_size, void* d_ws, size_t ws_size,
                              hipStream_t stream) {
    (void)in_sizes; (void)n_in; (void)d_ws; (void)ws_size; (void)out_size;
    // setup_inputs() order:
    //   0: nodes [B] int, 1: neighbors_1 [B,S] int, 2: neighbors_2 [B*S,S] int,
    //   3: node_features [N,D] f32, 4: W1 [H,2D] f32 (dead), 5: W2 [H,2D] f32,
    //   6: Wo [O,H] f32
    const int*   nodes  = (const int*)d_in[0];
    const int*   neigh1 = (const int*)d_in[1];
    const float* feats  = (const float*)d_in[3];
    const float* W2     = (const float*)d_in[5];
    const float* Wo     = (const float*)d_in[6];
    float*       out    = (float*)d_out;

    constexpr int B = 4096;
    dim3 grid(B / sage::TM);   // 256 blocks of 16 seed rows
    dim3 block(128);           // 4 wave32 waves
    graphsage_fused_kernel<<<grid, block, 0, stream>>>(
        nodes, neigh1, feats, W2, Wo, out);
}